// FDLayer_4612794876085
// MI455X (gfx1250) — compile-verified
//
#include <hip/hip_runtime.h>

// FDLayer synaptic-dynamics scan for gfx1250 (MI455X).
// B=32, T=2048, H=512. One thread per (b,h) channel; state (EPSC, Ca, R)
// lives in VGPRs. I_Ca streamed through LDS with double-buffered
// global_load_async_to_lds_b128 (ASYNCcnt path, CDNA5); non-temporal output
// stores. Substep recurrence strength-reduced to FMA cycles with exp2/rcp.

constexpr int B_ = 32;
constexpr int T_ = 2048;
constexpr int H_ = 512;
constexpr int HTILE  = 256;          // h-channels per block
constexpr int CH     = 32;           // time steps staged per LDS chunk
constexpr int NCHUNK = T_ / CH;      // 64
constexpr float LOG2E = 1.44269504088896340736f;

typedef __attribute__((address_space(3))) unsigned char lds_u8_t;

__device__ __forceinline__ void async_ld_b128(unsigned lds_off, const void* gaddr) {
    // GLOBAL_LOAD_ASYNC_TO_LDS_B128: vdst = wave-relative LDS byte offset,
    // vaddr = 64-bit global address (GV mode). Tracked by ASYNCcnt.
    asm volatile("global_load_async_to_lds_b128 %0, %1, off"
                 :: "v"(lds_off), "v"((unsigned long long)gaddr)
                 : "memory");
}

__device__ __forceinline__ void wait_async0() {
    asm volatile("s_wait_asynccnt 0x0" ::: "memory");
}

__device__ __forceinline__ float fast_exp2(float x) {
#if __has_builtin(__builtin_amdgcn_exp2f)
    return __builtin_amdgcn_exp2f(x);     // v_exp_f32
#else
    return __expf(x * 0.6931471805599453f);
#endif
}

__device__ __forceinline__ float fast_rcp(float x) {
#if __has_builtin(__builtin_amdgcn_rcpf)
    return __builtin_amdgcn_rcpf(x);      // v_rcp_f32 (~1 ulp)
#else
    return 1.0f / x;
#endif
}

struct Coef {
    float dt;
    float sX, sY;       // exp2 argument: x = fma(Ca, sX, sY)
    float pMax;         // P_rel_max
    float eA, eB;       // EPSC' = fma(EPSC, eA, -eB*temp)
    float kM, kD;       // kr = fma(sg, kD, kM)  (= dt*k_recov)
    float cA, cC, cAl;  // Ca' = fma(Ca, cA, cB_t); cB_t = fma(cAl, I_t, cC)
};

// One full time step (all substeps) on the three state registers.
template <int STEPS>
__device__ __forceinline__ void time_step(float& EPSC, float& Ca, float& R,
                                          float I_t, int steps, const Coef& k) {
    const float pI = k.pMax * I_t;            // per-t, off the substep chains
    const float cB = fmaf(k.cAl, I_t, k.cC);  // per-t Ca drive term

    auto body = [&]() {
        const float x    = fmaf(Ca, k.sX, k.sY);    // log2e*(Ca_mu-Ca)/sigma
        const float e    = fast_exp2(x);
        const float sg   = fast_rcp(1.0f + e);      // sigmoid
        const float temp = (pI * sg) * R;
        EPSC = fmaf(EPSC, k.eA, -(k.eB * temp));
        const float kr = fmaf(sg, k.kD, k.kM);      // dt*k_recov
        const float rB = fmaf(temp, -k.dt, kr);
        R  = fmaf(R, 1.0f - kr, rB);
        Ca = fmaf(Ca, k.cA, cB);
    };

    if (STEPS > 0) {
#pragma unroll
        for (int s = 0; s < STEPS; ++s) body();
    } else {
        for (int s = 0; s < steps; ++s) body();
    }
}

template <int STEPS>
__device__ __forceinline__ void scan_all(const float* gbase, float* obase,
                                         float (&buf)[2][CH][HTILE], unsigned lds0,
                                         int tid, int steps, const Coef& k,
                                         float EPSC, float Ca, float R) {
    // Loader mapping: each 1KB row (256 floats) = 64 lanes x 16B.
    const int sub = tid & 63;    // 16-byte group within row
    const int rq  = tid >> 6;    // row quarter: 0..3

    // ---- prime chunk 0 ----
#pragma unroll
    for (int p = 0; p < CH / 4; ++p) {
        const int r = rq * (CH / 4) + p;             // rows 0..CH-1
        async_ld_b128(lds0 + (unsigned)(r * HTILE * 4 + sub * 16),
                      gbase + (size_t)r * H_ + sub * 4);
    }
    wait_async0();
    __syncthreads();

    for (int c = 0; c < NCHUNK; ++c) {
        const int cur = c & 1;

        // ---- async prefetch of next chunk into the other buffer ----
        if (c + 1 < NCHUNK) {
            const unsigned lb = lds0 + (unsigned)(((c + 1) & 1) * CH * HTILE * 4);
            const float*   gb = gbase + (size_t)(c + 1) * CH * H_;
#pragma unroll
            for (int p = 0; p < CH / 4; ++p) {
                const int r = rq * (CH / 4) + p;
                async_ld_b128(lb + (unsigned)(r * HTILE * 4 + sub * 16),
                              gb + (size_t)r * H_ + sub * 4);
            }
        }

        // ---- recurrence over the staged chunk ----
#pragma unroll 4
        for (int r = 0; r < CH; ++r) {
            const float I_t = buf[cur][r][tid];
            time_step<STEPS>(EPSC, Ca, R, I_t, steps, k);
            __builtin_nontemporal_store(EPSC, obase + (size_t)(c * CH + r) * H_ + tid);
        }

        wait_async0();     // this wave's prefetch landed
        __syncthreads();   // whole block done reading 'cur'
    }
}

__global__ __launch_bounds__(HTILE) void fd_layer_scan_kernel(
    const float* __restrict__ I_Ca,
    const float* __restrict__ log_Ca_mu,
    const float* __restrict__ log_Ca_sigma,
    const float* __restrict__ log_tau_Ca,
    const float* __restrict__ log_alpha,
    const float* __restrict__ log_tau_EPSC,
    const float* __restrict__ log_beta,
    const float* __restrict__ presig_P,
    const float* __restrict__ log_k_min,
    const float* __restrict__ log_k_delta,
    const int*   __restrict__ ode_steps_p,
    float*       __restrict__ out)
{
    __shared__ float buf[2][CH][HTILE];   // 64 KB double buffer

    const int tid = threadIdx.x;
    const int b   = blockIdx.x >> 1;            // 2 h-tiles per batch row
    const int hb  = (blockIdx.x & 1) * HTILE;
    const int h   = hb + tid;

    const float* gbase = I_Ca + ((size_t)b * T_) * H_ + hb;  // + t*H_ selects row t
    float*       obase = out  + ((size_t)b * T_) * H_ + hb;

    const unsigned lds0 = (unsigned)(unsigned long long)(lds_u8_t*)&buf[0][0][0];

    // ---- per-channel parameter transforms (once) ----
    const int   steps     = *ode_steps_p;
    const float dt        = 1.0f / (float)steps;
    const float Ca_mu     = __expf(log_Ca_mu[h]);
    const float inv_sig   = __expf(-log_Ca_sigma[h]);
    const float inv_tauCa = __expf(-log_tau_Ca[h]);
    const float alpha     = __expf(log_alpha[h]);
    const float inv_tauE  = __expf(-log_tau_EPSC[h]);
    const float beta      = __expf(log_beta[h]);
    const float P_max     = 1.0f / (1.0f + __expf(-presig_P[h]));
    const float k_min     = __expf(log_k_min[h]);
    const float k_del     = __expf(log_k_delta[h]);

    Coef k;
    k.dt   = dt;
    k.sX   = -LOG2E * inv_sig;          // x = log2e*(Ca_mu - Ca)/sigma
    k.sY   =  LOG2E * inv_sig * Ca_mu;
    k.pMax = P_max;
    k.eA   = 1.0f - dt * inv_tauE;
    k.eB   = dt * beta;
    k.kM   = dt * k_min;
    k.kD   = dt * k_del;
    k.cA   = 1.0f - dt * inv_tauCa;
    k.cC   = dt * inv_tauCa * Ca_mu;
    k.cAl  = dt * alpha;

    const float EPSC0 = 0.0f, Ca0 = Ca_mu, R0 = 1.0f;

    if (steps == 4) {   // harness always uses ODE_STEPS = 4 -> unrolled body
        scan_all<4>(gbase, obase, buf, lds0, tid, steps, k, EPSC0, Ca0, R0);
    } else {            // generic fallback, runtime substep loop
        scan_all<0>(gbase, obase, buf, lds0, tid, steps, k, EPSC0, Ca0, R0);
    }
}

extern "C" void kernel_launch(void* const* d_in, const int* in_sizes, int n_in,
                              void* d_out, int out_size, void* d_ws, size_t ws_size,
                              hipStream_t stream) {
    (void)in_sizes; (void)n_in; (void)out_size; (void)d_ws; (void)ws_size;

    const float* I_Ca          = (const float*)d_in[0];
    const float* log_Ca_mu     = (const float*)d_in[1];
    const float* log_Ca_sigma  = (const float*)d_in[2];
    const float* log_tau_Ca    = (const float*)d_in[3];
    const float* log_alpha     = (const float*)d_in[4];
    const float* log_tau_EPSC  = (const float*)d_in[5];
    const float* log_beta      = (const float*)d_in[6];
    const float* presig_P      = (const float*)d_in[7];
    const float* log_k_min     = (const float*)d_in[8];
    const float* log_k_delta   = (const float*)d_in[9];
    const int*   ode_steps     = (const int*)d_in[10];
    float*       out           = (float*)d_out;

    const dim3 grid(B_ * (H_ / HTILE));   // 64 blocks
    const dim3 block(HTILE);              // 256 threads = 8 wave32

    fd_layer_scan_kernel<<<grid, block, 0, stream>>>(
        I_Ca, log_Ca_mu, log_Ca_sigma, log_tau_Ca, log_alpha, log_tau_EPSC,
        log_beta, presig_P, log_k_min, log_k_delta, ode_steps, out);
}